// VQmodulator_74509092651500
// MI455X (gfx1250) — compile-verified
//
#include <hip/hip_runtime.h>
#include <hip/hip_bf16.h>

// ---------------------------------------------------------------------------
// Problem constants (from reference)
// ---------------------------------------------------------------------------
#define BB    16
#define FF    256
#define CC    64
#define HW    4096           // 64*64
#define NE    512            // codebook size
#define NELEM 4194304        // total z elements
#define EPS   1e-5f

// Workspace layout (float offsets)
#define WS_SCALE 0           // 256  bn scale  (rstd*gamma)
#define WS_SHIFT 256         // 256  bn shift  (beta - mean*scale)
#define WS_E2    512         // 512  ||e_j||^2
#define WS_EN    1024        // 512  ||e_j||
#define WS_SC    1536        // 8 scalar accumulators
#define WS_ANG   2048        // 512*512 angles
#define WS_Z     264192      // 4194304 z buffer

// padded LDS row strides (floats): 16B-aligned rows, bank step 4 -> conflict-free
#define W1S 260              // row stride for W1 [64][256]
#define W2S 68               // row stride for W2 [64][64]
#define EMS 68               // row stride for emb [512][64]

typedef __attribute__((ext_vector_type(2))) float v2f;
typedef __attribute__((ext_vector_type(8))) float v8f;
typedef __attribute__((ext_vector_type(4))) int   i4;

typedef __attribute__((address_space(1))) i4* gp_b128;   // global int4*
typedef __attribute__((address_space(3))) i4* lp_b128;   // LDS int4*

__device__ __forceinline__ v8f wmma_f32(v2f a, v2f b, v8f c) {
    // D = A(16x4,f32) * B(4x16,f32) + C(16x16,f32)
    return __builtin_amdgcn_wmma_f32_16x16x4_f32(
        /*neg_a=*/false, a, /*neg_b=*/false, b,
        /*c_mod=*/(short)0, c, /*reuse_a=*/false, /*reuse_b=*/false);
}

// async 128-bit global->LDS copy (GLOBAL_LOAD_ASYNC_TO_LDS_B128, ASYNCcnt)
__device__ __forceinline__ void async_cp_b128(const float* g, float* l) {
    i4* gg = (i4*)const_cast<float*>(g);
    i4* ll = (i4*)l;
    __builtin_amdgcn_global_load_async_to_lds_b128(
        (gp_b128)gg, (lp_b128)ll, /*offset=*/0, /*cpol=*/0);
}
__device__ __forceinline__ void async_wait_all() {
    __builtin_amdgcn_s_wait_asynccnt(0);
}

// ---------------------------------------------------------------------------
// 0. zero scalar accumulators
// ---------------------------------------------------------------------------
__global__ void k_zero(float* sc) {
    if (threadIdx.x < 8) sc[threadIdx.x] = 0.0f;
}

// ---------------------------------------------------------------------------
// 1. BatchNorm statistics -> fused scale/shift per feature
// ---------------------------------------------------------------------------
__global__ __launch_bounds__(256) void k_bn_stats(const float* __restrict__ x,
                                                  const float* __restrict__ gamma,
                                                  const float* __restrict__ beta,
                                                  float* __restrict__ ws) {
    const int f = blockIdx.x;       // 0..255
    const int tid = threadIdx.x;
    __shared__ float ssum[256];
    __shared__ float ssq[256];
    float s = 0.f, q = 0.f;
    for (int b = 0; b < BB; ++b) {
        const float* p = x + ((size_t)b * FF + f) * HW;
        for (int i = tid; i < HW; i += 256) {
            float v = p[i];
            s += v; q += v * v;
        }
    }
    ssum[tid] = s; ssq[tid] = q;
    __syncthreads();
    for (int st = 128; st > 0; st >>= 1) {
        if (tid < st) { ssum[tid] += ssum[tid + st]; ssq[tid] += ssq[tid + st]; }
        __syncthreads();
    }
    if (tid == 0) {
        float mean = ssum[0] * (1.0f / 65536.0f);
        float var  = ssq[0]  * (1.0f / 65536.0f) - mean * mean;
        float rstd = rsqrtf(var + EPS);
        float sc   = rstd * gamma[f];
        ws[WS_SCALE + f] = sc;
        ws[WS_SHIFT + f] = beta[f] - mean * sc;
    }
}

// ---------------------------------------------------------------------------
// 2. codebook norms + hsw + clipped r (tiny)
// ---------------------------------------------------------------------------
__global__ void k_code_norms(const float* __restrict__ emb,
                             const float* __restrict__ r,
                             float* __restrict__ ws) {
    int j = blockIdx.x * blockDim.x + threadIdx.x;
    if (j < NE) {
        const float* e = emb + (size_t)j * CC;
        float s = 0.f;
        for (int k = 0; k < CC; ++k) { float v = e[k]; s += v * v; }
        ws[WS_E2 + j] = s;
        float en = sqrtf(s);
        ws[WS_EN + j] = en;
        float dr = r[j] - en;
        atomicAdd(&ws[WS_SC + 3], dr * dr);
        float rc = fminf(fmaxf(r[j], 0.9f), 1.1f);
        atomicAdd(&ws[WS_SC + 4], rc);
    }
}

// ---------------------------------------------------------------------------
// 3. fused BN-normalize + conv1(256->64) + conv2(64->64), WMMA f32 16x16x4
//    block = 256 threads = 8 waves; each wave owns a 16-token tile.
//    Weights staged to LDS via GLOBAL_LOAD_ASYNC_TO_LDS_B128 (row-major,
//    padded strides for alignment + conflict-free fragment reads).
// ---------------------------------------------------------------------------
__global__ __launch_bounds__(256) void k_fused_mlp(const float* __restrict__ x,
                                                   const float* __restrict__ w1,
                                                   const float* __restrict__ b1,
                                                   const float* __restrict__ w2,
                                                   const float* __restrict__ b2,
                                                   const float* __restrict__ ws,
                                                   float* __restrict__ zout) {
    __shared__ float s_w1[CC * W1S];      // [c][k], stride 260  (~66.6 KB)
    __shared__ float s_w2[CC * W2S];      // [o][k], stride 68   (~17.4 KB)
    __shared__ float s_scale[FF];
    __shared__ float s_shift[FF];
    __shared__ float s_b1[CC];
    __shared__ float s_b2[CC];
    __shared__ float s_yt[8][16 * 65];    // per-wave y tile, pad 65

    const int tid = threadIdx.x;
    // async stage W1: 64 rows x 64 b128-chunks = 4096 chunks, 16 per thread
    for (int ch = tid; ch < 4096; ch += 256) {
        int c = ch >> 6, q = (ch & 63) << 2;
        async_cp_b128(w1 + (size_t)c * FF + q, &s_w1[c * W1S + q]);
    }
    // async stage W2: 64 rows x 16 chunks = 1024 chunks, 4 per thread
    for (int ch = tid; ch < 1024; ch += 256) {
        int o = ch >> 4, q = (ch & 15) << 2;
        async_cp_b128(w2 + (size_t)o * CC + q, &s_w2[o * W2S + q]);
    }
    if (tid < FF) { s_scale[tid] = ws[WS_SCALE + tid]; s_shift[tid] = ws[WS_SHIFT + tid]; }
    if (tid < CC) { s_b1[tid] = b1[tid]; s_b2[tid] = b2[tid]; }
    async_wait_all();
    __syncthreads();

    const int wave = tid >> 5, lane = tid & 31;
    const int l15 = lane & 15;
    const int hi2 = (lane >= 16) ? 2 : 0;
    const int hi8 = (lane >= 16) ? 8 : 0;
    const int tok0 = (blockIdx.x * 8 + wave) * 16;
    const int b    = tok0 >> 12;          // token / 4096
    const int hw0  = tok0 & 4095;
    const float* xb = x + (size_t)b * FF * HW;
    float* yt = s_yt[wave];

    // ---- GEMM1: y[16x64] = bn(x)[16x256] * W1^T, K stepped by 4 ----
    v8f acc[4];
    #pragma unroll
    for (int nt = 0; nt < 4; ++nt) acc[nt] = v8f{};
    for (int k0 = 0; k0 < FF; k0 += 4) {
        int kk = k0 + hi2;
        float a0 = xb[(size_t)kk * HW + hw0 + l15];
        float a1 = xb[(size_t)(kk + 1) * HW + hw0 + l15];
        a0 = a0 * s_scale[kk]     + s_shift[kk];
        a1 = a1 * s_scale[kk + 1] + s_shift[kk + 1];
        v2f a = {a0, a1};
        #pragma unroll
        for (int nt = 0; nt < 4; ++nt) {
            const float* wrow = &s_w1[(nt * 16 + l15) * W1S + kk];
            v2f bf = { wrow[0], wrow[1] };
            acc[nt] = wmma_f32(a, bf, acc[nt]);
        }
    }
    // + bias, spill to LDS in [m][c] layout for re-fragmenting
    #pragma unroll
    for (int nt = 0; nt < 4; ++nt) {
        int n = nt * 16 + l15;
        float bias = s_b1[n];
        #pragma unroll
        for (int rr = 0; rr < 8; ++rr)
            yt[(rr + hi8) * 65 + n] = acc[nt][rr] + bias;
    }

    // ---- GEMM2: z[16x64] = y[16x64] * W2^T ----
    v8f zacc[4];
    #pragma unroll
    for (int nt = 0; nt < 4; ++nt) zacc[nt] = v8f{};
    for (int k0 = 0; k0 < CC; k0 += 4) {
        int kk = k0 + hi2;
        v2f a = { yt[l15 * 65 + kk], yt[l15 * 65 + kk + 1] };
        #pragma unroll
        for (int nt = 0; nt < 4; ++nt) {
            const float* wrow = &s_w2[(nt * 16 + l15) * W2S + kk];
            v2f bf = { wrow[0], wrow[1] };
            zacc[nt] = wmma_f32(a, bf, zacc[nt]);
        }
    }
    // + bias, bounce through LDS, then channel-strided store into [B,C,H,W] layout
    #pragma unroll
    for (int nt = 0; nt < 4; ++nt) {
        int n = nt * 16 + l15;
        float bias = s_b2[n];
        #pragma unroll
        for (int rr = 0; rr < 8; ++rr)
            yt[(rr + hi8) * 65 + n] = zacc[nt][rr] + bias;
    }
    const size_t zbase = (size_t)b * CC * HW;
    for (int i = 0; i < 32; ++i) {
        int e = i * 32 + lane;           // 16*64 = 1024 elements
        int m = e >> 6, c = e & 63;
        zout[zbase + (size_t)c * HW + hw0 + m] = yt[m * 65 + c];
    }
}

// ---------------------------------------------------------------------------
// 4. VQ: distances via WMMA, argmin, gather z_q, accumulate sum of min dists
//    Embedding staged to LDS via async B128 copies (row-major, stride 68).
// ---------------------------------------------------------------------------
__global__ __launch_bounds__(256) void k_vq(const float* __restrict__ z,
                                            const float* __restrict__ emb,
                                            const float* __restrict__ e2g,
                                            float* __restrict__ sum_sq,
                                            float* __restrict__ outq) {
    __shared__ float s_emb[NE * EMS];     // [j][k], stride 68  (~139 KB)
    __shared__ float s_e2[NE];
    __shared__ float s_bv[8][256];
    __shared__ int   s_bi[8][256];
    __shared__ int   s_widx[8][16];

    const int tid = threadIdx.x;
    // 512 rows x 16 b128-chunks = 8192 chunks, 32 per thread
    for (int ch = tid; ch < 8192; ch += 256) {
        int j = ch >> 4, q = (ch & 15) << 2;
        async_cp_b128(emb + (size_t)j * CC + q, &s_emb[j * EMS + q]);
    }
    for (int i = tid; i < NE; i += 256) s_e2[i] = e2g[i];
    async_wait_all();
    __syncthreads();

    const int wave = tid >> 5, lane = tid & 31;
    const int l15 = lane & 15;
    const int hi2 = (lane >= 16) ? 2 : 0;
    const int hi8 = (lane >= 16) ? 8 : 0;
    const int row0 = (blockIdx.x * 8 + wave) * 16;
    const float* zr = z + (size_t)row0 * CC;

    // Hoist all A fragments (16 rows x K=64) once
    v2f afr[16];
    #pragma unroll
    for (int ks = 0; ks < 16; ++ks) {
        int kk = ks * 4 + hi2;
        afr[ks] = { zr[l15 * CC + kk], zr[l15 * CC + kk + 1] };
    }

    float bestv[8];
    int   besti[8];
    #pragma unroll
    for (int rr = 0; rr < 8; ++rr) { bestv[rr] = 3.4e38f; besti[rr] = 0; }

    for (int j0 = 0; j0 < NE; j0 += 16) {
        const float* erow = &s_emb[(j0 + l15) * EMS];
        v8f acc = v8f{};
        #pragma unroll
        for (int ks = 0; ks < 16; ++ks) {
            int kk = ks * 4 + hi2;
            v2f bf = { erow[kk], erow[kk + 1] };
            acc = wmma_f32(afr[ks], bf, acc);
        }
        float e2j = s_e2[j0 + l15];
        #pragma unroll
        for (int rr = 0; rr < 8; ++rr) {
            float key = e2j - 2.0f * acc[rr];   // d - ||z||^2 (row-constant)
            if (key < bestv[rr]) { bestv[rr] = key; besti[rr] = j0 + l15; }
        }
    }
    #pragma unroll
    for (int rr = 0; rr < 8; ++rr) {
        s_bv[wave][(rr + hi8) * 16 + l15] = bestv[rr];
        s_bi[wave][(rr + hi8) * 16 + l15] = besti[rr];
    }
    // per-wave LDS region; DS ops are in-order within the wave
    if (lane < 16) {
        int m = lane;
        float bv = 3.4e38f; int bj = 0;
        for (int c = 0; c < 16; ++c) {
            float v = s_bv[wave][m * 16 + c];
            if (v < bv) { bv = v; bj = s_bi[wave][m * 16 + c]; }
        }
        s_widx[wave][m] = bj;
        float z2 = 0.f;
        const float* zm = zr + m * CC;
        for (int k = 0; k < CC; ++k) { float v = zm[k]; z2 += v * v; }
        atomicAdd(sum_sq, z2 + bv);            // min ||z_row - e||^2
    }
    // gather z_q rows (straight-through output == embedding[idx])
    for (int i = 0; i < 32; ++i) {
        int e = i * 32 + lane;                 // 16*64 elements
        int m = e >> 6, k = e & 63;
        int code = s_widx[wave][m];
        outq[(size_t)(row0 + m) * CC + k] = s_emb[code * EMS + k];
    }
}

// ---------------------------------------------------------------------------
// 5. codebook angle matrix
// ---------------------------------------------------------------------------
__global__ __launch_bounds__(256) void k_ang(const float* __restrict__ emb,
                                             const float* __restrict__ ws,
                                             float* __restrict__ ang) {
    int t = blockIdx.x * 256 + threadIdx.x;    // 512*512 entries
    int i = t >> 9, j = t & 511;
    const float* ei = emb + (size_t)i * CC;
    const float* ej = emb + (size_t)j * CC;
    float dot = 0.f;
    for (int k = 0; k < CC; ++k) dot += ei[k] * ej[k];
    float c = dot / (ws[WS_EN + i] * ws[WS_EN + j]);
    c = fminf(fmaxf(c, -0.99999f), 0.99999f);
    ang[t] = acosf(c);
}

// 6. per-row unbiased variance -> accumulate
__global__ __launch_bounds__(256) void k_rowvar(const float* __restrict__ ang,
                                                float* __restrict__ sum_var) {
    int i = blockIdx.x, tid = threadIdx.x;
    __shared__ float ss[256];
    __shared__ float sq[256];
    float a0 = ang[(size_t)i * NE + tid];
    float a1 = ang[(size_t)i * NE + tid + 256];
    ss[tid] = a0 + a1; sq[tid] = a0 * a0 + a1 * a1;
    __syncthreads();
    for (int st = 128; st > 0; st >>= 1) {
        if (tid < st) { ss[tid] += ss[tid + st]; sq[tid] += sq[tid + st]; }
        __syncthreads();
    }
    if (tid == 0) {
        float s = ss[0], q = sq[0];
        float var = (q - s * s * (1.0f / 512.0f)) * (1.0f / 511.0f);
        atomicAdd(sum_var, var);
    }
}

// 7. per-column second-smallest -> accumulate
__global__ __launch_bounds__(256) void k_colmin(const float* __restrict__ ang,
                                                float* __restrict__ sum_second) {
    int j = blockIdx.x, tid = threadIdx.x;
    __shared__ float s1[256];
    __shared__ float s2[256];
    float a = ang[(size_t)tid * NE + j];
    float b = ang[(size_t)(tid + 256) * NE + j];
    s1[tid] = fminf(a, b); s2[tid] = fmaxf(a, b);
    __syncthreads();
    for (int st = 128; st > 0; st >>= 1) {
        if (tid < st) {
            float a1 = s1[tid], a2 = s2[tid];
            float b1 = s1[tid + st], b2 = s2[tid + st];
            s1[tid] = fminf(a1, b1);
            s2[tid] = fminf(fmaxf(a1, b1), fminf(a2, b2));
        }
        __syncthreads();
    }
    if (tid == 0) atomicAdd(sum_second, s2[0]);
}

// 8. finalize scalars
__global__ void k_final(const float* __restrict__ ws, float* __restrict__ out) {
    float sum_sq  = ws[WS_SC + 0];
    float sum_var = ws[WS_SC + 1];
    float sum_sec = ws[WS_SC + 2];
    float sum_hsw = ws[WS_SC + 3];
    float sum_rc  = ws[WS_SC + 4];
    float hsw   = sum_hsw * (1.0f / 512.0f);
    float cbvar = sum_var * (1.0f / 512.0f);
    float tmd   = sum_sec * (1.0f / 512.0f);
    float mse   = sum_sq * (1.0f / (float)NELEM);
    float loss  = 2.0f * mse + hsw + (cbvar - tmd);   // (1+BETA)*mse + ...
    out[0] = loss;
    out[1 + NELEM + 0] = cbvar;
    out[1 + NELEM + 1] = tmd;
    out[1 + NELEM + 2] = hsw;
    out[1 + NELEM + 3] = 0.0f;                        // cb_loss
    out[1 + NELEM + 4] = sum_rc * (1.0f / 512.0f);    // mean_r
}

// ---------------------------------------------------------------------------
extern "C" void kernel_launch(void* const* d_in, const int* in_sizes, int n_in,
                              void* d_out, int out_size, void* d_ws, size_t ws_size,
                              hipStream_t stream) {
    const float* x    = (const float*)d_in[0];
    const float* g    = (const float*)d_in[1];
    const float* be   = (const float*)d_in[2];
    const float* w1   = (const float*)d_in[3];
    const float* b1   = (const float*)d_in[4];
    const float* w2   = (const float*)d_in[5];
    const float* b2   = (const float*)d_in[6];
    const float* emb  = (const float*)d_in[7];
    const float* r    = (const float*)d_in[8];
    float* out = (float*)d_out;
    float* ws  = (float*)d_ws;
    float* zbuf = ws + WS_Z;
    float* angm = ws + WS_ANG;

    k_zero      <<<1,    32, 0, stream>>>(ws + WS_SC);
    k_bn_stats  <<<256, 256, 0, stream>>>(x, g, be, ws);
    k_code_norms<<<2,   256, 0, stream>>>(emb, r, ws);
    k_fused_mlp <<<512, 256, 0, stream>>>(x, w1, b1, w2, b2, ws, zbuf);
    k_vq        <<<512, 256, 0, stream>>>(zbuf, emb, ws + WS_E2, ws + WS_SC, out + 1);
    k_ang       <<<1024,256, 0, stream>>>(emb, ws, angm);
    k_rowvar    <<<512, 256, 0, stream>>>(angm, ws + WS_SC + 1);
    k_colmin    <<<512, 256, 0, stream>>>(angm, ws + WS_SC + 2);
    k_final     <<<1,     1, 0, stream>>>(ws, out);
}